// XRDModel_21741124452355
// MI455X (gfx1250) — compile-verified
//
#include <hip/hip_runtime.h>
#include <math.h>

#define B_SI_C 4.1491f
#define B_O_C  5.803f
#define CUTOFF_C 3.9f
#define PI_F 3.14159265358979323846f

typedef __attribute__((ext_vector_type(2))) float v2f;
typedef __attribute__((ext_vector_type(8))) float v8f;

// ---------------------------------------------------------------------------
// Kernel 1: setup — mean scattering length, 3x3 cell inverses, |det|,
// zero the histogram accumulator, zero the padded 16 x kmax A matrix,
// and build the padded r array (pad = 1.0) for the branchless WMMA loop.
// ---------------------------------------------------------------------------
__global__ void setup_kernel(const float* __restrict__ cell,
                             const int* __restrict__ an,
                             const float* __restrict__ r_bins,
                             float* __restrict__ ws_bmean,
                             float* __restrict__ ws_inv,
                             float* __restrict__ ws_vol,
                             float* __restrict__ ws_hist,
                             float* __restrict__ ws_rpad,
                             float* __restrict__ ws_Apad,
                             int S, int N, int NR, int kmax) {
  __shared__ float red[256];
  int tid = threadIdx.x;
  float acc = 0.f;
  for (int i = tid; i < N; i += 256) acc += (an[i] == 14) ? B_SI_C : B_O_C;
  red[tid] = acc;
  __syncthreads();
  for (int w = 128; w > 0; w >>= 1) {
    if (tid < w) red[tid] += red[tid + w];
    __syncthreads();
  }
  if (tid == 0) ws_bmean[0] = red[0] / (float)N;

  if (tid < S) {
    const float* c = cell + tid * 9;
    float a00=c[0],a01=c[1],a02=c[2],a10=c[3],a11=c[4],a12=c[5],a20=c[6],a21=c[7],a22=c[8];
    float det = a00*(a11*a22 - a12*a21) - a01*(a10*a22 - a12*a20) + a02*(a10*a21 - a11*a20);
    float id = 1.f / det;
    float* inv = ws_inv + tid * 9;
    inv[0] =  (a11*a22 - a12*a21)*id; inv[1] = -(a01*a22 - a02*a21)*id; inv[2] =  (a01*a12 - a02*a11)*id;
    inv[3] = -(a10*a22 - a12*a20)*id; inv[4] =  (a00*a22 - a02*a20)*id; inv[5] = -(a00*a12 - a02*a10)*id;
    inv[6] =  (a10*a21 - a11*a20)*id; inv[7] = -(a00*a21 - a01*a20)*id; inv[8] =  (a00*a11 - a01*a10)*id;
    ws_vol[tid] = fabsf(det);
  }
  for (int i = tid; i < S * NR; i += 256) ws_hist[i] = 0.f;
  for (int i = tid; i < kmax; i += 256) ws_rpad[i] = (i < NR) ? r_bins[i] : 1.f;
  for (int i = tid; i < 16 * kmax; i += 256) ws_Apad[i] = 0.f;
}

// ---------------------------------------------------------------------------
// Kernel 2: Gaussian-smeared weighted pair histogram.
// One block per (system, i-atom); threads stride over j. Windowed scatter:
// beyond 12 sigma = 6 bins the Gaussian is < 6e-32 (below fp32 noise).
// LDS histogram (ds_add_f32), then global_atomic_add_f32 per block.
// ---------------------------------------------------------------------------
__global__ void hist_kernel(const float* __restrict__ pos,
                            const float* __restrict__ cell,
                            const int* __restrict__ an,
                            const float* __restrict__ r_bins,
                            const float* __restrict__ ws_bmean,
                            const float* __restrict__ ws_inv,
                            float* __restrict__ ws_hist,
                            int S, int N, int NR) {
  extern __shared__ float lhist[];
  int bid = blockIdx.x;
  int s = bid / N;
  int i = bid % N;
  int tid = threadIdx.x;
  for (int r = tid; r < NR; r += blockDim.x) lhist[r] = 0.f;
  __syncthreads();

  const float* ps = pos + (size_t)s * N * 3;
  float xi = ps[i*3+0], yi = ps[i*3+1], zi = ps[i*3+2];
  const float* cl = cell + s * 9;
  const float* iv = ws_inv + s * 9;
  float c00=cl[0],c01=cl[1],c02=cl[2],c10=cl[3],c11=cl[4],c12=cl[5],c20=cl[6],c21=cl[7],c22=cl[8];
  float i00=iv[0],i01=iv[1],i02=iv[2],i10=iv[3],i11=iv[4],i12=iv[5],i20=iv[6],i21=iv[7],i22=iv[8];

  float bm = ws_bmean[0];
  float wscale = (((an[i] == 14) ? B_SI_C : B_O_C)) / (bm * bm);
  float r0 = r_bins[0];
  float dr = r_bins[1] - r0;
  float inv_dr = 1.f / dr;
  float inv_sig = 2.f / dr;          // sigma = dr/2

  for (int j = tid; j < N; j += blockDim.x) {
    int jn = j + blockDim.x;
    if (jn < N) __builtin_prefetch(&ps[jn * 3], 0, 0);   // global_prefetch_b8
    if (j == i) continue;
    float dx = ps[j*3+0] - xi;
    float dy = ps[j*3+1] - yi;
    float dz = ps[j*3+2] - zi;
    // frac = disp @ inv_cell (row-vector convention)
    float fx = dx*i00 + dy*i10 + dz*i20;
    float fy = dx*i01 + dy*i11 + dz*i21;
    float fz = dx*i02 + dy*i12 + dz*i22;
    fx -= rintf(fx); fy -= rintf(fy); fz -= rintf(fz);
    float mx = fx*c00 + fy*c10 + fz*c20;
    float my = fx*c01 + fy*c11 + fz*c21;
    float mz = fx*c02 + fy*c12 + fz*c22;
    float dist = sqrtf(mx*mx + my*my + mz*mz + 1e-12f);

    float w = wscale * ((an[j] == 14) ? B_SI_C : B_O_C);
    float center = (dist - r0) * inv_dr;
    int lo = (int)ceilf(center - 6.f);  if (lo < 0) lo = 0;
    int hi = (int)floorf(center + 6.f); if (hi > NR - 1) hi = NR - 1;
    for (int r = lo; r <= hi; ++r) {
      float x = (dist - (r0 + (float)r * dr)) * inv_sig;
      atomicAdd(&lhist[r], w * __expf(-0.5f * x * x));
    }
  }
  __syncthreads();
  for (int r = tid; r < NR; r += blockDim.x) {
    float v = lhist[r];
    if (v != 0.f) atomicAdd(&ws_hist[s * NR + r], v);
  }
}

// ---------------------------------------------------------------------------
// Kernel 3: finalize G(r), T(r); write A = 4*pi*rho*dr*(G-1)*r^2 into the
// zero-padded 16 x kmax matrix consumed by the WMMA S(Q) kernel.
// ---------------------------------------------------------------------------
__global__ void finalize_kernel(const float* __restrict__ r_bins,
                                const float* __restrict__ ws_hist,
                                const float* __restrict__ ws_vol,
                                float* __restrict__ ws_Apad,
                                float* __restrict__ out_G,
                                float* __restrict__ out_T,
                                int S, int N, int NR, int kmax) {
  int idx = blockIdx.x * blockDim.x + threadIdx.x;
  if (idx >= S * NR) return;
  int s = idx / NR, r = idx % NR;
  float rb = r_bins[r];
  float dr = r_bins[1] - r_bins[0];
  float sigma = 0.5f * dr;
  float kern = dr / (sigma * sqrtf(2.f * PI_F));
  float V = ws_vol[s];
  float h = ws_hist[idx] * kern;
  float G = h * V / ((float)N * (float)N * 4.f * PI_F * rb * rb * dr);
  float rho = (float)N / V;
  out_G[idx] = G;
  out_T[idx] = 4.f * PI_F * rho * rb * G;
  ws_Apad[s * kmax + r] = 4.f * PI_F * rho * dr * (G - 1.f) * rb * rb;
}

// ---------------------------------------------------------------------------
// Kernel 4: S(Q) = 1 + A[16 x kmax] x sinc[kmax x NQ] via V_WMMA_F32_16X16X4_F32.
// One wave per 16-wide q tile; K stepped by 4. Fully branchless inner loop:
// A is zero-padded, r is pad-1.0, and sinc uses v_sin_f32 * v_rcp_f32.
// A layout (32-bit A 16x4): lanes 0-15 hold K=0(v0),1(v1); lanes 16-31 K=2,3.
// B layout mirrors the K split; C/D: vgpr v -> M = v + 8*(lane>=16), N = lane&15.
// ---------------------------------------------------------------------------
__global__ void sq_wmma_kernel(const float* __restrict__ q_bins,
                               const float* __restrict__ ws_rpad,
                               const float* __restrict__ ws_Apad,
                               float* __restrict__ out_S,
                               int S, int NQ, int kmax) {
  int lane = threadIdx.x;            // 0..31 (wave32)
  int half = lane >> 4;              // 0: K=0,1 ; 1: K=2,3
  int sub  = lane & 15;
  int tile = blockIdx.x;
  int n = tile * 16 + sub;
  float qv = (n < NQ) ? q_bins[n] : 1.f;
  const float* arow = ws_Apad + sub * kmax + half * 2;

  v8f acc = {};
  for (int k0 = 0; k0 < kmax; k0 += 4) {
    v2f a, b;
    a.x = arow[k0];
    a.y = arow[k0 + 1];
    float ra = ws_rpad[k0 + half * 2];
    float rb = ws_rpad[k0 + half * 2 + 1];
    float qra = qv * ra, qrb = qv * rb;
    b.x = __sinf(qra) * __builtin_amdgcn_rcpf(qra);
    b.y = __sinf(qrb) * __builtin_amdgcn_rcpf(qrb);
    acc = __builtin_amdgcn_wmma_f32_16x16x4_f32(
        /*neg_a=*/false, a, /*neg_b=*/false, b,
        /*c_mod=*/(short)0, acc, /*reuse_a=*/false, /*reuse_b=*/false);
  }
  if (n < NQ) {
#pragma unroll
    for (int v = 0; v < 8; ++v) {
      int m = v + half * 8;
      if (m < S) out_S[m * NQ + n] = 1.f + acc[v];
    }
  }
}

// ---------------------------------------------------------------------------
// Kernel 5: tetrahedral order parameter (last system). One thread per Si;
// O positions staged in LDS (24 KB << 320 KB/WGP); register top-4 insertion.
// ---------------------------------------------------------------------------
__global__ void qtet_kernel(const float* __restrict__ pos_last,
                            const float* __restrict__ cell_last,
                            const float* __restrict__ inv_last,
                            float* __restrict__ out_q,
                            int N, int n_si) {
  extern __shared__ float opos[];
  int nO = N - n_si;
  int tid = threadIdx.x;
  for (int idx = tid; idx < nO * 3; idx += blockDim.x)
    opos[idx] = pos_last[n_si * 3 + idx];
  __syncthreads();

  int c = blockIdx.x * blockDim.x + tid;
  if (c >= n_si) return;

  float xi = pos_last[c*3+0], yi = pos_last[c*3+1], zi = pos_last[c*3+2];
  float c00=cell_last[0],c01=cell_last[1],c02=cell_last[2];
  float c10=cell_last[3],c11=cell_last[4],c12=cell_last[5];
  float c20=cell_last[6],c21=cell_last[7],c22=cell_last[8];
  float i00=inv_last[0],i01=inv_last[1],i02=inv_last[2];
  float i10=inv_last[3],i11=inv_last[4],i12=inv_last[5];
  float i20=inv_last[6],i21=inv_last[7],i22=inv_last[8];

  float d0=1e30f,d1=1e30f,d2=1e30f,d3=1e30f;
  float v0x=0,v0y=0,v0z=0, v1x=0,v1y=0,v1z=0, v2x=0,v2y=0,v2z=0, v3x=0,v3y=0,v3z=0;
  int cnt = 0;

  for (int o = 0; o < nO; ++o) {
    float dx = opos[o*3+0] - xi;
    float dy = opos[o*3+1] - yi;
    float dz = opos[o*3+2] - zi;
    float fx = dx*i00 + dy*i10 + dz*i20;
    float fy = dx*i01 + dy*i11 + dz*i21;
    float fz = dx*i02 + dy*i12 + dz*i22;
    fx -= rintf(fx); fy -= rintf(fy); fz -= rintf(fz);
    float mx = fx*c00 + fy*c10 + fz*c20;
    float my = fx*c01 + fy*c11 + fz*c21;
    float mz = fx*c02 + fy*c12 + fz*c22;
    float dd = sqrtf(mx*mx + my*my + mz*mz + 1e-12f);
    if (dd <= CUTOFF_C) {
      cnt++;
      if (dd < d3) {
        if (dd < d2) {
          d3 = d2; v3x = v2x; v3y = v2y; v3z = v2z;
          if (dd < d1) {
            d2 = d1; v2x = v1x; v2y = v1y; v2z = v1z;
            if (dd < d0) {
              d1 = d0; v1x = v0x; v1y = v0y; v1z = v0z;
              d0 = dd; v0x = mx; v0y = my; v0z = mz;
            } else { d1 = dd; v1x = mx; v1y = my; v1z = mz; }
          } else { d2 = dd; v2x = mx; v2y = my; v2z = mz; }
        } else { d3 = dd; v3x = mx; v3y = my; v3z = mz; }
      }
    }
  }

  float q = 0.f;
  if (cnt >= 4) {
    float n0 = rsqrtf(v0x*v0x + v0y*v0y + v0z*v0z + 1e-12f);
    float n1 = rsqrtf(v1x*v1x + v1y*v1y + v1z*v1z + 1e-12f);
    float n2 = rsqrtf(v2x*v2x + v2y*v2y + v2z*v2z + 1e-12f);
    float n3 = rsqrtf(v3x*v3x + v3y*v3y + v3z*v3z + 1e-12f);
    float u[4][3] = {
      {v0x*n0, v0y*n0, v0z*n0}, {v1x*n1, v1y*n1, v1z*n1},
      {v2x*n2, v2y*n2, v2z*n2}, {v3x*n3, v3y*n3, v3z*n3}};
    const int jj[6] = {0,0,0,1,1,2};
    const int kk[6] = {1,2,3,2,3,3};
    float sum = 0.f;
#pragma unroll
    for (int p = 0; p < 6; ++p) {
      float cs = u[jj[p]][0]*u[kk[p]][0] + u[jj[p]][1]*u[kk[p]][1] + u[jj[p]][2]*u[kk[p]][2];
      cs = fminf(1.f, fmaxf(-1.f, cs));
      float t = cs + (1.f/3.f);
      sum += t * t;
    }
    q = 1.f - 0.375f * sum;
  }
  out_q[c] = q;
}

// ---------------------------------------------------------------------------
extern "C" void kernel_launch(void* const* d_in, const int* in_sizes, int n_in,
                              void* d_out, int out_size, void* d_ws, size_t ws_size,
                              hipStream_t stream) {
  const float* pos    = (const float*)d_in[0];
  const float* cell   = (const float*)d_in[1];
  const float* r_bins = (const float*)d_in[2];
  const float* q_bins = (const float*)d_in[3];
  const int*   an     = (const int*)d_in[4];

  int NR = in_sizes[2];
  int NQ = in_sizes[3];
  int N  = in_sizes[4];
  int S  = in_sizes[1] / 9;
  int n_si = N / 3;
  int nO = N - n_si;
  int kmax = (NR + 3) & ~3;

  float* ws       = (float*)d_ws;
  float* ws_bmean = ws;                       // 1 (+pad)
  float* ws_inv   = ws + 4;                   // 9*S
  float* ws_vol   = ws_inv + 9 * S;           // S
  float* ws_hist  = ws_vol + S;               // S*NR
  float* ws_rpad  = ws_hist + S * NR;         // kmax
  float* ws_Apad  = ws_rpad + kmax;           // 16*kmax (zero-padded)

  float* out   = (float*)d_out;
  float* out_G = out;                         // S*NR
  float* out_T = out + S * NR;                // S*NR
  float* out_S = out + 2 * S * NR;            // S*NQ
  float* out_q = out_S + S * NQ;              // n_si

  setup_kernel<<<1, 256, 0, stream>>>(cell, an, r_bins, ws_bmean, ws_inv,
                                      ws_vol, ws_hist, ws_rpad, ws_Apad,
                                      S, N, NR, kmax);

  hist_kernel<<<S * N, 256, NR * sizeof(float), stream>>>(
      pos, cell, an, r_bins, ws_bmean, ws_inv, ws_hist, S, N, NR);

  int tot = S * NR;
  finalize_kernel<<<(tot + 255) / 256, 256, 0, stream>>>(
      r_bins, ws_hist, ws_vol, ws_Apad, out_G, out_T, S, N, NR, kmax);

  sq_wmma_kernel<<<(NQ + 15) / 16, 32, 0, stream>>>(
      q_bins, ws_rpad, ws_Apad, out_S, S, NQ, kmax);

  qtet_kernel<<<(n_si + 255) / 256, 256, (size_t)nO * 3 * sizeof(float), stream>>>(
      pos + (size_t)(S - 1) * N * 3, cell + (size_t)(S - 1) * 9,
      ws_inv + (size_t)(S - 1) * 9, out_q, N, n_si);
}